// countRes_45191645888950
// MI455X (gfx1250) — compile-verified
//
#include <hip/hip_runtime.h>

#define THREADS 256
#define NBINS   10
#define NCOUNT  (2 * NBINS)   // [0..9] = ano (gt > 0.5), [10..19] = nor

__global__ void zero_kernel(float* out, int n) {
  int i = blockIdx.x * blockDim.x + threadIdx.x;
  if (i < n) out[i] = 0.0f;
}

// One element -> one conflict-free LDS atomic.
// hist layout: hist[counter * THREADS + tid]; counter stride (256 ints) is a
// multiple of the 64 LDS banks, so bank = tid % 64: no conflicts in a wave32.
__device__ __forceinline__ void bump(int* hist, int tid, float r, float g) {
  // Faithful to reference: truncate-toward-zero of the correctly rounded
  // IEEE quotient r / 0.1f (NOT r * 10.0f, which differs at bin boundaries).
  int bin = (int)(r / 0.1f);
  bin = min(max(bin, 0), NBINS - 1);          // clip(, 0, 9) -> v_med3_i32
  int idx = (g > 0.5f) ? bin : (bin + NBINS); // ano first, matching return order
  atomicAdd(&hist[idx * THREADS + tid], 1);   // ds_add_u32 (no return needed)
}

__global__ __launch_bounds__(THREADS) void hist_kernel(const float* __restrict__ res,
                                                       const float* __restrict__ gt,
                                                       float* __restrict__ out,
                                                       int n) {
  __shared__ int hist[NCOUNT * THREADS];      // 20 KB: per-thread-private counters
  const int tid = threadIdx.x;
#pragma unroll
  for (int c = 0; c < NCOUNT; ++c) hist[c * THREADS + tid] = 0;
  __syncthreads();

  const int n4 = n >> 2;
  const float4* __restrict__ r4 = reinterpret_cast<const float4*>(res);
  const float4* __restrict__ g4 = reinterpret_cast<const float4*>(gt);
  const int stride = gridDim.x * blockDim.x;

  for (int i = blockIdx.x * blockDim.x + tid; i < n4; i += stride) {
    // Warm L2 one grid-sweep ahead (global_prefetch_b8). Working set (134 MB)
    // fits in the 192 MB L2, so keep default RT temporal hints on the loads.
    if (i + stride < n4) {
      __builtin_prefetch(&r4[i + stride], 0, 1);
      __builtin_prefetch(&g4[i + stride], 0, 1);
    }
    float4 r = r4[i];   // global_load_b128
    float4 g = g4[i];   // global_load_b128
    bump(hist, tid, r.x, g.x);
    bump(hist, tid, r.y, g.y);
    bump(hist, tid, r.z, g.z);
    bump(hist, tid, r.w, g.w);
  }

  // Tail (n not divisible by 4): handled by block 0 only.
  const int base = n4 << 2;
  if (blockIdx.x == 0 && tid < (n - base)) {
    bump(hist, tid, res[base + tid], gt[base + tid]);
  }

  __syncthreads();

  // Tree-reduce the 256 per-thread partials for each of the 20 counters.
  for (int s = THREADS / 2; s > 0; s >>= 1) {
    if (tid < s) {
#pragma unroll
      for (int c = 0; c < NCOUNT; ++c)
        hist[c * THREADS + tid] += hist[c * THREADS + tid + s];
    }
    __syncthreads();
  }

  // Counts are integers <= 2^24: float accumulation is exact and
  // order-independent -> deterministic across graph replays.
  if (tid < NCOUNT) atomicAdd(&out[tid], (float)hist[tid * THREADS]);
}

extern "C" void kernel_launch(void* const* d_in, const int* in_sizes, int n_in,
                              void* d_out, int out_size, void* d_ws, size_t ws_size,
                              hipStream_t stream) {
  const float* res = (const float*)d_in[0];  // resmap: 64*512*512 f32
  const float* gt  = (const float*)d_in[1];  // gtmap:  64*1*512*512 f32
  float* out = (float*)d_out;                // 20 counts (ano[10] ++ nor[10])
  const int n = in_sizes[0];

  // d_out is poisoned by the harness; zero it before atomic accumulation.
  zero_kernel<<<(out_size + 255) / 256, 256, 0, stream>>>(out, out_size);

  int n4 = n >> 2;
  int blocks = (n4 + THREADS - 1) / THREADS;
  if (blocks > 4096) blocks = 4096;   // grid-stride: ~4 float4 iters/thread
  if (blocks < 1) blocks = 1;
  hist_kernel<<<blocks, THREADS, 0, stream>>>(res, gt, out, n);
}